// HOGGenerator_20126216749686
// MI455X (gfx1250) — compile-verified
//
#include <hip/hip_runtime.h>
#include <math.h>

// ---------------- problem constants (from reference) ----------------
#define BATCH_CH   3
#define IMG_H      224
#define IMG_W      224
#define NBINS      9
#define POOLSZ     8
#define HC         28          // IMG_H / POOLSZ
#define WC         28
#define GW         16          // gaussian kernel size, std = 8
// ---------------- tiling ----------------
#define TILE       32          // 4x4 cells of 8x8 pixels
#define HALO       34          // TILE + 1 halo each side
#define LPITCH     36          // padded LDS pitch (floats) to dodge bank conflicts
#define NCELL      16          // 4x4 cells per tile
#define NWAVE      8
#define TPB        256         // 8 wave32s

#ifndef __has_builtin
#define __has_builtin(x) 0
#endif

#define AS1 __attribute__((address_space(1)))
#define AS3 __attribute__((address_space(3)))

typedef __attribute__((ext_vector_type(2))) float v2f;
typedef __attribute__((ext_vector_type(8))) float v8f;

#if __has_builtin(__builtin_amdgcn_global_load_async_to_lds_b32)
#define ASYNC_VIA_BUILTIN 1
#else
#define ASYNC_VIA_BUILTIN 0
#endif

__device__ __forceinline__ void async_copy_b32(const float* gsrc, float* ldst) {
#if ASYNC_VIA_BUILTIN
    __builtin_amdgcn_global_load_async_to_lds_b32(
        (AS1 int*)gsrc, (AS3 int*)ldst, /*offset=*/0, /*cpol=*/0);
#else
    // GV mode: VDST = LDS byte address, VADDR = 64-bit global address.
    unsigned int laddr = (unsigned int)(unsigned long long)(AS3 float*)ldst;
    unsigned long long gaddr = (unsigned long long)(size_t)gsrc;
    asm volatile("global_load_async_to_lds_b32 %0, %1, off"
                 :: "v"(laddr), "v"(gaddr) : "memory");
#endif
}

__device__ __forceinline__ void wait_asynccnt0() {
#if __has_builtin(__builtin_amdgcn_s_wait_asynccnt)
    __builtin_amdgcn_s_wait_asynccnt(0);
#else
    asm volatile("s_wait_asynccnt 0" ::: "memory");
#endif
}

__global__ __launch_bounds__(TPB)
void hog_cdna5_kernel(const float* __restrict__ x, float* __restrict__ out)
{
    __shared__ float tile[HALO * LPITCH];        // 34x34 reflect-padded pixels
    __shared__ float cpart[NWAVE][NCELL * 16];   // per-wave 16x16 WMMA accumulators
    __shared__ float hist16[NCELL * 16];         // 16 cells x 16 bins (9 used)
    __shared__ float cinv[NCELL];                // per-cell 1/L2norm

    const int t    = threadIdx.x;
    const int lane = t & 31;
    const int w    = t >> 5;
    const int tx = blockIdx.x % (IMG_W / TILE);  // 0..6
    const int ty = blockIdx.x / (IMG_W / TILE);  // 0..6
    const int c  = blockIdx.y;
    const int b  = blockIdx.z;

    const int x0 = tx * TILE;
    const int y0 = ty * TILE;

    const float* plane = x + ((size_t)b * BATCH_CH + c) * (size_t)(IMG_H * IMG_W);

    // -------- stage reflect-padded 34x34 halo tile into LDS (async path) --------
    for (int i = t; i < HALO * HALO; i += TPB) {
        int ly = i / HALO;
        int lx = i - ly * HALO;
        int gy = y0 - 1 + ly;
        int gx = x0 - 1 + lx;
        gy = (gy < 0) ? -gy : ((gy >= IMG_H) ? 2 * IMG_H - 2 - gy : gy);
        gx = (gx < 0) ? -gx : ((gx >= IMG_W) ? 2 * IMG_W - 2 - gx : gx);
        async_copy_b32(plane + (size_t)gy * IMG_W + gx, &tile[ly * LPITCH + lx]);
    }

    // -------- gaussian kernel normalization (overlaps async copy): sum(k2)==S^2 --------
    float S = 0.0f;
#pragma unroll
    for (int i = 0; i < GW; ++i) {
        float n = ((float)i - 7.5f) * 0.125f;    // (i - mean) / std, std = 8
        S += expf(-0.5f * n * n);
    }
    const float invS2    = 1.0f / (S * S);
    const float binScale = (float)NBINS / 3.14159265358979323846f;

    wait_asynccnt0();
    __syncthreads();

    // -------- Sobel + bin + gaussian-weighted magnitude; pool via WMMA --------
    // Hist[16 cells][16 bins] = sum_K A[cell][k] * B[k][bin], A = cell one-hot,
    // B = bin one-hot * mag  (exactly the reference's one_hot()*mag + sum-pool).
    // Wave w handles rows 4w..4w+3; group g = one full row, lane = column.
    v8f acc = {0.0f, 0.0f, 0.0f, 0.0f, 0.0f, 0.0f, 0.0f, 0.0f};
    const int n16 = lane & 15;

#pragma unroll
    for (int g = 0; g < 4; ++g) {
        const int py = w * 4 + g;
        const int px = lane;
        const float* r0 = &tile[py * LPITCH + px];   // halo rows py, py+1, py+2
        const float* r1 = r0 + LPITCH;
        const float* r2 = r1 + LPITCH;
        float a00 = r0[0], a01 = r0[1], a02 = r0[2];
        float a10 = r1[0],              a12 = r1[2];
        float a20 = r2[0], a21 = r2[1], a22 = r2[2];
        // cross-correlation with wx = [[1,0,-1],[2,0,-2],[1,0,-1]] and wx^T
        float gxv = (a00 - a02) + 2.0f * (a10 - a12) + (a20 - a22);
        float gyv = (a00 + 2.0f * a01 + a02) - (a20 + 2.0f * a21 + a22);
        float nrm = sqrtf(gxv * gxv + gyv * gyv);
        float phase = atan2f(gxv, gyv) * binScale;   // note: atan2(gx, gy) as in source
        int bin = (int)floorf(phase);
        bin %= NBINS; if (bin < 0) bin += NBINS;
        float ny = ((float)((y0 + py) & (GW - 1)) - 7.5f) * 0.125f;
        float nx = ((float)((x0 + px) & (GW - 1)) - 7.5f) * 0.125f;
        float mag = nrm * expf(-0.5f * (nx * nx + ny * ny)) * invS2;

        const int cellrow = py >> 3;                 // uniform within the row
        // 8 K-chunks of 4 pixels; each chunk lies entirely in one cell.
        #pragma unroll
        for (int j = 0; j < 8; ++j) {
            const int cell_j = cellrow * 4 + (j >> 1);
            // A fragment: K-independent -> same value in both VGPRs / lane halves.
            float av = (n16 == cell_j) ? 1.0f : 0.0f;
            v2f afrag = {av, av};
            // B fragment: lane half picks pixels {0,1} vs {2,3} of the chunk.
            int src = 4 * j + ((lane >> 4) << 1);
            float m0 = __shfl(mag, src, 32);
            int   b0 = __shfl(bin, src, 32);
            float m1 = __shfl(mag, src + 1, 32);
            int   b1 = __shfl(bin, src + 1, 32);
            v2f bfrag;
            bfrag.x = (b0 == n16) ? m0 : 0.0f;
            bfrag.y = (b1 == n16) ? m1 : 0.0f;
            acc = __builtin_amdgcn_wmma_f32_16x16x4_f32(
                false, afrag, false, bfrag, (short)0, acc, false, false);
        }
    }

    // -------- spill per-wave C (16x16) to LDS: lane<16 -> M=r, lane>=16 -> M=r+8 --------
    {
        const int mbase = (lane >> 4) << 3;
#pragma unroll
        for (int r = 0; r < 8; ++r)
            cpart[w][(mbase + r) * 16 + n16] = acc[r];
    }
    __syncthreads();

    // -------- cross-wave reduction: hist16[cell][bin] --------
    {
        float s = 0.0f;
#pragma unroll
        for (int wv = 0; wv < NWAVE; ++wv) s += cpart[wv][t];
        hist16[t] = s;
    }
    __syncthreads();

    // -------- per-cell L2 normalization over the 9 bins --------
    if (t < NCELL) {
        float ss = 0.0f;
#pragma unroll
        for (int k = 0; k < NBINS; ++k) {
            float v = hist16[t * 16 + k];
            ss += v * v;
        }
        cinv[t] = 1.0f / fmaxf(sqrtf(ss), 1e-12f);
    }
    __syncthreads();

    // -------- write out[b][(hc*28+wc)][c*9+bin] --------
    if (t < NCELL * NBINS) {
        int cell = t / NBINS;
        int bin  = t - cell * NBINS;
        int cy = cell >> 2, cx = cell & 3;
        int row = (ty * 4 + cy) * WC + (tx * 4 + cx);
        size_t oidx = ((size_t)b * (HC * WC) + row) * (BATCH_CH * NBINS)
                    + (size_t)(c * NBINS + bin);
        out[oidx] = hist16[cell * 16 + bin] * cinv[cell];
    }
}

extern "C" void kernel_launch(void* const* d_in, const int* in_sizes, int n_in,
                              void* d_out, int out_size, void* d_ws, size_t ws_size,
                              hipStream_t stream) {
    (void)n_in; (void)out_size; (void)d_ws; (void)ws_size;
    const float* x = (const float*)d_in[0];
    float* out = (float*)d_out;
    int B = in_sizes[0] / (BATCH_CH * IMG_H * IMG_W);   // 64
    dim3 grid((IMG_H / TILE) * (IMG_W / TILE), BATCH_CH, B);  // (49, 3, 64)
    hipLaunchKernelGGL(hog_cdna5_kernel, grid, dim3(TPB), 0, stream, x, out);
}